// GCN_scheduling_87797721465439
// MI455X (gfx1250) — compile-verified
//
#include <hip/hip_runtime.h>
#include <math.h>

#define NSLOPE 0.2f
#define H 128
#define FIN 5

typedef float v2f __attribute__((ext_vector_type(2)));
typedef float v8f __attribute__((ext_vector_type(8)));

// Monotone unsigned encoding of IEEE-754 float for integer atomicMax.
__device__ __forceinline__ unsigned fkey(float f) {
    unsigned u = __float_as_uint(f);
    return (u & 0x80000000u) ? ~u : (u | 0x80000000u);
}
__device__ __forceinline__ float funkey(unsigned k) {
    unsigned u = (k & 0x80000000u) ? (k & 0x7FFFFFFFu) : ~k;
    return __uint_as_float(u);
}

__device__ __forceinline__ void edge_sd(const int* __restrict__ ei, int E, long e,
                                        int& s, int& d) {
    if (e < (long)E) { s = ei[e]; d = ei[(long)E + e]; }
    else             { s = d = (int)(e - E); }
}

// ---------------------------------------------------------------------------
// Kernel 1: h1[N,128] = x[N,5] @ W1[5,128] via V_WMMA_F32_16X16X4_F32.
// One wave handles a 16-node x 128-feature tile: 8 col-blocks x 2 K-chunks.
// A (16x4 f32): lanes 0-15 hold K={0,1}, lanes 16-31 hold K={2,3} (VGPR0/VGPR1).
// The K=4 "padding" lanes use an unconditional in-bounds load times a lane
// mask instead of a divergent guard, so the inner loop is branch-free.
// ---------------------------------------------------------------------------
__global__ __launch_bounds__(256) void k_gemm1(const float* __restrict__ x,
                                               const float* __restrict__ W,
                                               float* __restrict__ h1, int n) {
    int gwave = (int)((blockIdx.x * blockDim.x + threadIdx.x) >> 5);
    int lane  = threadIdx.x & 31;
    int half  = lane >> 4;     // 0: lanes 0-15, 1: lanes 16-31
    int m     = lane & 15;
    long base = (long)gwave * 16;
    if (base >= n) return;                 // uniform across the wave
    bool full = (base + 16 <= (long)n);    // uniform: whole 16-row tile valid
    float msk = (half == 0) ? 1.f : 0.f;   // lanes 16-31 carry K=6,7 -> zero

    long row = base + m;
    v2f a0, a1;
    if (full) {
        // K-chunk 0 covers K=0..3 (all valid, FIN=5)
        a0.x = x[row * FIN + 2 * half + 0];
        a0.y = x[row * FIN + 2 * half + 1];
        // K-chunk 1 covers K=4..7 -> only K=4 real; in-bounds load * mask
        a1.x = x[row * FIN + 4] * msk;
        a1.y = 0.f;
    } else {
        bool rv = row < (long)n;
        a0.x = rv ? x[row * FIN + 2 * half + 0] : 0.f;
        a0.y = rv ? x[row * FIN + 2 * half + 1] : 0.f;
        a1.x = rv ? x[row * FIN + 4] * msk : 0.f;
        a1.y = 0.f;
    }

    for (int b = 0; b < 8; ++b) {
        int ncol = b * 16 + m;
        v2f b0, b1v;
        b0.x  = W[(2 * half + 0) * H + ncol];
        b0.y  = W[(2 * half + 1) * H + ncol];
        b1v.x = W[4 * H + ncol] * msk;       // index <= 639, always in-bounds
        b1v.y = 0.f;

        v8f c = {0.f, 0.f, 0.f, 0.f, 0.f, 0.f, 0.f, 0.f};
        c = __builtin_amdgcn_wmma_f32_16x16x4_f32(false, a0, false, b0,
                                                  (short)0, c, false, false);
        c = __builtin_amdgcn_wmma_f32_16x16x4_f32(false, a1, false, b1v,
                                                  (short)0, c, false, false);

        float* hp = h1 + (base + 8 * half) * H + ncol;   // row (base+8*half), col ncol
        if (full) {
#pragma unroll
            for (int r = 0; r < 8; ++r) hp[(size_t)r * H] = c[r];   // offsets 0,512,...
        } else {
#pragma unroll
            for (int r = 0; r < 8; ++r) {
                long nd = base + r + 8 * half;
                if (nd < (long)n) h1[nd * H + ncol] = c[r];
            }
        }
    }
}

// ---------------------------------------------------------------------------
// Kernel 2: per-node attention scalars for layer 1 + init of softmax state.
// One wave per node; float4 per lane; shuffle reduction (wave32).
// ---------------------------------------------------------------------------
__global__ __launch_bounds__(256) void k_node1(const float* __restrict__ h1,
                                               const float* __restrict__ a_s,
                                               const float* __restrict__ a_d,
                                               float* __restrict__ as1,
                                               float* __restrict__ ad1,
                                               unsigned* __restrict__ m1,
                                               float* __restrict__ s1,
                                               float* __restrict__ out1, int n) {
    long node = (long)((blockIdx.x * blockDim.x + threadIdx.x) >> 5);
    int lane  = threadIdx.x & 31;
    if (node >= n) return;
    float4 hv = ((const float4*)(h1 + node * H))[lane];
    float4 av = ((const float4*)a_s)[lane];
    float4 dv = ((const float4*)a_d)[lane];
    float ps = hv.x * av.x + hv.y * av.y + hv.z * av.z + hv.w * av.w;
    float pd = hv.x * dv.x + hv.y * dv.y + hv.z * dv.z + hv.w * dv.w;
#pragma unroll
    for (int o = 16; o > 0; o >>= 1) {
        ps += __shfl_xor(ps, o, 32);
        pd += __shfl_xor(pd, o, 32);
    }
    float4 z = {0.f, 0.f, 0.f, 0.f};
    ((float4*)(out1 + node * H))[lane] = z;   // zero aggregation target
    if (lane == 0) {
        as1[node] = ps;
        ad1[node] = pd;
        m1[node]  = fkey(-__builtin_inff());
        s1[node]  = 0.f;
    }
}

// ---------------------------------------------------------------------------
// Edge pass A (shared by both layers): segment max of leaky-relu logits.
// ---------------------------------------------------------------------------
__global__ void k_emax(const int* __restrict__ ei, int E, int M,
                       const float* __restrict__ as, const float* __restrict__ ad,
                       unsigned* __restrict__ mm) {
    long e = (long)blockIdx.x * blockDim.x + threadIdx.x;
    if (e >= M) return;
    int s, d; edge_sd(ei, E, e, s, d);
    float l = as[s] + ad[d];
    l = l > 0.f ? l : NSLOPE * l;
    atomicMax(&mm[d], fkey(l));
}

// ---------------------------------------------------------------------------
// Edge pass B (shared): segment sum of exp(logit - max).
// ---------------------------------------------------------------------------
__global__ void k_esum(const int* __restrict__ ei, int E, int M,
                       const float* __restrict__ as, const float* __restrict__ ad,
                       const unsigned* __restrict__ mm, float* __restrict__ ss) {
    long e = (long)blockIdx.x * blockDim.x + threadIdx.x;
    if (e >= M) return;
    int s, d; edge_sd(ei, E, e, s, d);
    float l = as[s] + ad[d];
    l = l > 0.f ? l : NSLOPE * l;
    atomicAdd(&ss[d], expf(l - funkey(mm[d])));
}

// ---------------------------------------------------------------------------
// Kernel 5: heavy layer-1 scatter. One wave per edge; 128 floats (float4/lane)
// gathered from h1[src], weighted, atomically accumulated into out1[dst].
// ---------------------------------------------------------------------------
__global__ __launch_bounds__(256) void k_eagg1(const int* __restrict__ ei, int E, int M,
                                               const float* __restrict__ as,
                                               const float* __restrict__ ad,
                                               const unsigned* __restrict__ mm,
                                               const float* __restrict__ ss,
                                               const float* __restrict__ h1,
                                               float* __restrict__ out1) {
    long e   = (long)((blockIdx.x * (long)blockDim.x + threadIdx.x) >> 5);
    int lane = threadIdx.x & 31;
    if (e >= M) return;
    int s, d; edge_sd(ei, E, e, s, d);
    float l = as[s] + ad[d];
    l = l > 0.f ? l : NSLOPE * l;
    float w = expf(l - funkey(mm[d])) / (ss[d] + 1e-16f);
    float4 hv = ((const float4*)(h1 + (long)s * H))[lane];
    float* op = out1 + (long)d * H + lane * 4;
    atomicAdd(op + 0, w * hv.x);
    atomicAdd(op + 1, w * hv.y);
    atomicAdd(op + 2, w * hv.z);
    atomicAdd(op + 3, w * hv.w);
}

// ---------------------------------------------------------------------------
// Kernel 6: x2 = elu(out1 + b1); h2 = x2 @ W2; layer-2 scalars + state init.
// ---------------------------------------------------------------------------
__global__ __launch_bounds__(256) void k_node2(const float* __restrict__ out1,
                                               const float* __restrict__ b1,
                                               const float* __restrict__ W2,
                                               const float* __restrict__ a_s2,
                                               const float* __restrict__ a_d2,
                                               float* __restrict__ h2,
                                               float* __restrict__ as2,
                                               float* __restrict__ ad2,
                                               unsigned* __restrict__ m2,
                                               float* __restrict__ s2,
                                               float* __restrict__ out2, int n) {
    long node = (long)((blockIdx.x * blockDim.x + threadIdx.x) >> 5);
    int lane  = threadIdx.x & 31;
    if (node >= n) return;
    float4 v  = ((const float4*)(out1 + node * H))[lane];
    float4 bv = ((const float4*)b1)[lane];
    v.x += bv.x; v.y += bv.y; v.z += bv.z; v.w += bv.w;
    v.x = v.x > 0.f ? v.x : expf(v.x) - 1.f;
    v.y = v.y > 0.f ? v.y : expf(v.y) - 1.f;
    v.z = v.z > 0.f ? v.z : expf(v.z) - 1.f;
    v.w = v.w > 0.f ? v.w : expf(v.w) - 1.f;
    float4 wv = ((const float4*)W2)[lane];
    float p = v.x * wv.x + v.y * wv.y + v.z * wv.z + v.w * wv.w;
#pragma unroll
    for (int o = 16; o > 0; o >>= 1) p += __shfl_xor(p, o, 32);
    if (lane == 0) {
        h2[node]   = p;
        as2[node]  = p * a_s2[0];
        ad2[node]  = p * a_d2[0];
        m2[node]   = fkey(-__builtin_inff());
        s2[node]   = 0.f;
        out2[node] = 0.f;
    }
}

// ---------------------------------------------------------------------------
// Kernel 9: layer-2 scalar scatter.
// ---------------------------------------------------------------------------
__global__ void k_eagg2(const int* __restrict__ ei, int E, int M,
                        const float* __restrict__ as, const float* __restrict__ ad,
                        const unsigned* __restrict__ mm, const float* __restrict__ ss,
                        const float* __restrict__ h2, float* __restrict__ out2) {
    long e = (long)blockIdx.x * blockDim.x + threadIdx.x;
    if (e >= M) return;
    int s, d; edge_sd(ei, E, e, s, d);
    float l = as[s] + ad[d];
    l = l > 0.f ? l : NSLOPE * l;
    float w = expf(l - funkey(mm[d])) / (ss[d] + 1e-16f);
    atomicAdd(&out2[d], w * h2[s]);
}

// ---------------------------------------------------------------------------
// Kernel 10: sigmoid((out2 + b2) * lin_w)
// ---------------------------------------------------------------------------
__global__ void k_final(const float* __restrict__ out2, const float* __restrict__ b2,
                        const float* __restrict__ lin_w, float* __restrict__ out, int n) {
    long i = (long)blockIdx.x * blockDim.x + threadIdx.x;
    if (i >= n) return;
    float v = (out2[i] + b2[0]) * lin_w[0];
    out[i] = 1.f / (1.f + expf(-v));
}

extern "C" void kernel_launch(void* const* d_in, const int* in_sizes, int n_in,
                              void* d_out, int out_size, void* d_ws, size_t ws_size,
                              hipStream_t stream) {
    (void)n_in; (void)out_size; (void)ws_size;
    const float* x      = (const float*)d_in[0];
    const int*   ei     = (const int*)d_in[1];
    const float* W1     = (const float*)d_in[2];
    const float* a_src1 = (const float*)d_in[3];
    const float* a_dst1 = (const float*)d_in[4];
    const float* b1     = (const float*)d_in[5];
    const float* W2     = (const float*)d_in[6];
    const float* a_src2 = (const float*)d_in[7];
    const float* a_dst2 = (const float*)d_in[8];
    const float* b2     = (const float*)d_in[9];
    const float* lin_w  = (const float*)d_in[10];
    float* out = (float*)d_out;

    const int n = in_sizes[0] / FIN;
    const int E = in_sizes[1] / 2;
    const int M = E + n;   // edges + self-loops

    // Workspace layout (floats)
    float* h1   = (float*)d_ws;
    float* out1 = h1 + (size_t)n * H;
    float* as1  = out1 + (size_t)n * H;
    float* ad1  = as1 + n;
    float* s1   = ad1 + n;
    unsigned* m1 = (unsigned*)(s1 + n);
    float* h2   = (float*)(m1 + n);
    float* as2  = h2 + n;
    float* ad2  = as2 + n;
    float* s2   = ad2 + n;
    float* out2 = s2 + n;
    unsigned* m2 = (unsigned*)(out2 + n);

    const int TB = 256;
    int waves  = (n + 15) / 16;
    int gblk   = (waves + 7) / 8;             // 8 waves per 256-thread block
    int nwblk  = (n + 7) / 8;                 // one wave per node
    int eblk   = (M + TB - 1) / TB;           // one thread per edge
    int ewblk  = (int)(((long)M + 7) / 8);    // one wave per edge

    // Layer 1
    k_gemm1<<<gblk, TB, 0, stream>>>(x, W1, h1, n);
    k_node1<<<nwblk, TB, 0, stream>>>(h1, a_src1, a_dst1, as1, ad1, m1, s1, out1, n);
    k_emax <<<eblk, TB, 0, stream>>>(ei, E, M, as1, ad1, m1);
    k_esum <<<eblk, TB, 0, stream>>>(ei, E, M, as1, ad1, m1, s1);
    k_eagg1<<<ewblk, TB, 0, stream>>>(ei, E, M, as1, ad1, m1, s1, h1, out1);

    // ELU + layer-2 projection/scalars
    k_node2<<<nwblk, TB, 0, stream>>>(out1, b1, W2, a_src2, a_dst2,
                                      h2, as2, ad2, m2, s2, out2, n);

    // Layer 2 (scalar features)
    k_emax <<<eblk, TB, 0, stream>>>(ei, E, M, as2, ad2, m2);
    k_esum <<<eblk, TB, 0, stream>>>(ei, E, M, as2, ad2, m2, s2);
    k_eagg2<<<eblk, TB, 0, stream>>>(ei, E, M, as2, ad2, m2, s2, h2, out2);

    // Final linear + sigmoid
    k_final<<<(n + TB - 1) / TB, TB, 0, stream>>>(out2, b2, lin_w, out, n);
}